// FlameMeshSE3Extraction_2963527434718
// MI455X (gfx1250) — compile-verified
//
#include <hip/hip_runtime.h>
#include <hip/hip_bf16.h>
#include <stdint.h>

// Problem constants (match the reference).
#define T_FRAMES 128
#define N_VERT   5023
#define K_NBR    16

// ---- ext-vector types for the gfx1250 TDM builtin (6-arg clang-23 form) ----
typedef __attribute__((ext_vector_type(4))) unsigned int u32x4;
typedef __attribute__((ext_vector_type(8))) int          i32x8;
typedef __attribute__((ext_vector_type(4))) int          i32x4;

// Generic (flat) pointers to LDS carry the LDS byte offset in their low 32
// bits (SHARED_BASE aperture lives in the high bits) -> D#.lds_addr.
__device__ __forceinline__ unsigned lds_offset(const void* p) {
  return (unsigned)(uintptr_t)p;
}

// One TDM DMA: copy `nelem` contiguous dwords from global -> LDS.
// D# per CDNA5 ISA ch.8: group0 = {count/flags, lds_addr, global_addr, type=2},
// group1 = {data_size=4B, tensor_dim0=nelem, tensor_dim1=1, tile_dim0=nelem,
//           tile_dim1=0 (1-D), stride0=nelem}. Groups 2/3 (and the extra
//           group of the 6-arg builtin) zero: tensor is <=2-D.
__device__ __forceinline__ void tdm_load_row(const float* gsrc,
                                             unsigned lds_byte,
                                             unsigned nelem) {
  unsigned long long ga = (unsigned long long)(uintptr_t)gsrc;
  u32x4 g0;
  g0[0] = 1u;                                                  // count=1, user D#
  g0[1] = lds_byte;                                            // lds_addr[31:0]
  g0[2] = (unsigned)(ga & 0xffffffffull);                      // global_addr[31:0]
  g0[3] = (unsigned)((ga >> 32) & 0x01ffffffull) | (2u << 30); // addr[56:32]|type=2
  i32x8 g1;
  g1[0] = (int)(2u << 16);                                     // data_size = 4 bytes
  g1[1] = (int)((nelem & 0xffffu) << 16);                      // tensor_dim0[15:0]
  g1[2] = (int)(((nelem >> 16) & 0xffffu) | (1u << 16));       // dim0[31:16]|tensor_dim1=1
  g1[3] = (int)((nelem & 0xffffu) << 16);                      // tile_dim0 = nelem
  g1[4] = 0;                                                   // tile_dim1=0, tile_dim2=0
  g1[5] = (int)nelem;                                          // tensor_dim0_stride
  g1[6] = 0;
  g1[7] = 0;
  i32x4 z4 = {0, 0, 0, 0};
  i32x8 z8 = {0, 0, 0, 0, 0, 0, 0, 0};
  __builtin_amdgcn_tensor_load_to_lds(g0, g1, z4, z4, z8, 0);
}

// 256 threads = 8 wave32s. Each block: one time frame, one 1024-vertex chunk.
// Dynamic LDS: canonical frame (15072 f32) + deformed frame (15072 f32).
__global__ __launch_bounds__(256) void flame_se3_kernel(
    const float* __restrict__ can,   // [T,V,3]
    const float* __restrict__ def,   // [T,V,3]
    const int* __restrict__ nbr,     // [V,K] int32
    const unsigned char* __restrict__ msk, // [V,K] bool (1 byte each)
    float* __restrict__ out) {       // [T*V*4] quat then [T*V*3] trans
  constexpr int NVF    = N_VERT * 3;   // 15069 floats per frame per array
  constexpr int SC_PAD = 15072;        // 16B-aligned region size
  constexpr int CHUNKS = 5;            // ceil(5023/1024)
  constexpr int CHUNK  = 1024;
  constexpr int VPT    = 4;            // vertices per thread

  extern __shared__ float smem[];
  float* sC = smem;            // canonical[t] staged
  float* sD = smem + SC_PAD;   // deformed[t] staged

  const int t = (int)blockIdx.x / CHUNKS;
  const int c = (int)blockIdx.x % CHUNKS;

  // Wave 0 DMAs the whole frame into LDS via the Tensor Data Mover and waits
  // on its own TENSORcnt; the rest of the workgroup waits at the barrier.
  if (threadIdx.x < 32u) {
    tdm_load_row(can + (size_t)t * NVF, lds_offset(sC), (unsigned)NVF);
    tdm_load_row(def + (size_t)t * NVF, lds_offset(sD), (unsigned)NVF);
    __builtin_amdgcn_s_wait_tensorcnt(0);
  }
  __syncthreads();

  // Per-vertex Horn N-matrix state for VPT vertices (kept in registers so the
  // serial power-iteration chain gets VPT-way ILP below).
  float A00[VPT], A01[VPT], A02[VPT], A03[VPT], A11[VPT], A12[VPT], A13[VPT],
        A22[VPT], A23[VPT], A33[VPT];
  float D0[VPT], D1[VPT], D2[VPT], D3[VPT];
  float QW[VPT], QX[VPT], QY[VPT], QZ[VPT];
  float TX[VPT], TY[VPT], TZ[VPT];
  int   VID[VPT];

  const int v0 = c * CHUNK + (int)threadIdx.x;

  // ---- Phase 1: gather + 3x3 covariance + Horn N-matrix, per vertex ----
#pragma unroll
  for (int rep = 0; rep < VPT; ++rep) {
    const int v = v0 + rep * 256;
    VID[rep] = v;
    const int vc = (v < N_VERT) ? v : (N_VERT - 1);  // clamp tail (store skipped)

    const float cvx = sC[3 * vc + 0], cvy = sC[3 * vc + 1], cvz = sC[3 * vc + 2];
    const float dvx = sD[3 * vc + 0], dvy = sD[3 * vc + 1], dvz = sD[3 * vc + 2];
    TX[rep] = dvx - cvx; TY[rep] = dvy - cvy; TZ[rep] = dvz - cvz;

    // Neighbor indices: 64B coalesced; mask: 16B coalesced.
    int nb[K_NBR];
    {
      const int4* nb4 = (const int4*)(nbr + (size_t)vc * K_NBR);
#pragma unroll
      for (int q = 0; q < 4; ++q) {
        int4 r = nb4[q];
        nb[4 * q + 0] = r.x; nb[4 * q + 1] = r.y;
        nb[4 * q + 2] = r.z; nb[4 * q + 3] = r.w;
      }
    }
    const uint4 mm = *(const uint4*)(msk + (size_t)vc * K_NBR);
    const unsigned mw[4] = {mm.x, mm.y, mm.z, mm.w};

    // cov[i][j] = sum_k (m*c_nb[i] - c_v[i]) * (m*d_nb[j] - d_v[j])
    // (masked slots still subtract the center, matching the reference exactly).
    float c00 = 0.f, c01 = 0.f, c02 = 0.f;
    float c10 = 0.f, c11 = 0.f, c12 = 0.f;
    float c20 = 0.f, c21 = 0.f, c22 = 0.f;
#pragma unroll
    for (int k = 0; k < K_NBR; ++k) {
      const int j = nb[k];
      const float m = ((mw[k >> 2] >> ((k & 3) * 8)) & 0xffu) ? 1.0f : 0.0f;
      const float px = __builtin_fmaf(m, sC[3 * j + 0], -cvx);
      const float py = __builtin_fmaf(m, sC[3 * j + 1], -cvy);
      const float pz = __builtin_fmaf(m, sC[3 * j + 2], -cvz);
      const float qx = __builtin_fmaf(m, sD[3 * j + 0], -dvx);
      const float qy = __builtin_fmaf(m, sD[3 * j + 1], -dvy);
      const float qz = __builtin_fmaf(m, sD[3 * j + 2], -dvz);
      c00 = __builtin_fmaf(px, qx, c00);
      c01 = __builtin_fmaf(px, qy, c01);
      c02 = __builtin_fmaf(px, qz, c02);
      c10 = __builtin_fmaf(py, qx, c10);
      c11 = __builtin_fmaf(py, qy, c11);
      c12 = __builtin_fmaf(py, qz, c12);
      c20 = __builtin_fmaf(pz, qx, c20);
      c21 = __builtin_fmaf(pz, qy, c21);
      c22 = __builtin_fmaf(pz, qz, c22);
    }

    // Horn's 4x4 symmetric N: its max-eigenvalue eigenvector is the quaternion
    // of the proper Kabsch rotation (== reference's SVD + det correction).
    const float a00 =  c00 + c11 + c22;
    const float a01 =  c12 - c21;
    const float a02 =  c20 - c02;
    const float a03 =  c01 - c10;
    const float a11 =  c00 - c11 - c22;
    const float a12 =  c01 + c10;
    const float a13 =  c20 + c02;
    const float a22 = -c00 + c11 - c22;
    const float a23 =  c12 + c21;
    const float a33 = -c00 - c11 + c22;
    A00[rep] = a00; A01[rep] = a01; A02[rep] = a02; A03[rep] = a03;
    A11[rep] = a11; A12[rep] = a12; A13[rep] = a13;
    A22[rep] = a22; A23[rep] = a23; A33[rep] = a33;

    // Gershgorin shift so the max eigenvalue of (N + shift*I) is dominant.
    const float r0 = fabsf(a00) + fabsf(a01) + fabsf(a02) + fabsf(a03);
    const float r1 = fabsf(a01) + fabsf(a11) + fabsf(a12) + fabsf(a13);
    const float r2 = fabsf(a02) + fabsf(a12) + fabsf(a22) + fabsf(a23);
    const float r3 = fabsf(a03) + fabsf(a13) + fabsf(a23) + fabsf(a33);
    const float sh = fmaxf(fmaxf(r0, r1), fmaxf(r2, r3)) + 1e-6f;
    D0[rep] = a00 + sh; D1[rep] = a11 + sh;
    D2[rep] = a22 + sh; D3[rep] = a33 + sh;

    QW[rep] = 1.0f; QX[rep] = 0.31f; QY[rep] = 0.22f; QZ[rep] = 0.13f;
  }

  // ---- Phase 2: one power-iteration loop advancing VPT quaternions per
  // iteration (VPT-way ILP inside the serial dot->rsqrt->scale chain) ----
  for (int it = 0; it < 24; ++it) {
#pragma unroll
    for (int r = 0; r < VPT; ++r) {
      const float rw = D0[r] * QW[r] + A01[r] * QX[r] + A02[r] * QY[r] + A03[r] * QZ[r];
      const float rx = A01[r] * QW[r] + D1[r] * QX[r] + A12[r] * QY[r] + A13[r] * QZ[r];
      const float ry = A02[r] * QW[r] + A12[r] * QX[r] + D2[r] * QY[r] + A23[r] * QZ[r];
      const float rz = A03[r] * QW[r] + A13[r] * QX[r] + A23[r] * QY[r] + D3[r] * QZ[r];
      const float inv = rsqrtf(__builtin_fmaf(rw, rw,
                         __builtin_fmaf(rx, rx,
                         __builtin_fmaf(ry, ry, rz * rz))) + 1e-20f);
      QW[r] = rw * inv; QX[r] = rx * inv; QY[r] = ry * inv; QZ[r] = rz * inv;
    }
  }

  // ---- Phase 3: stores (quat 16B-aligned float4; translation 3 floats) ----
#pragma unroll
  for (int r = 0; r < VPT; ++r) {
    const int v = VID[r];
    if (v >= N_VERT) continue;
    const size_t idx = (size_t)t * N_VERT + (size_t)v;
    ((float4*)out)[idx] = make_float4(QW[r], QX[r], QY[r], QZ[r]);
    float* tr = out + (size_t)T_FRAMES * N_VERT * 4 + idx * 3;
    tr[0] = TX[r];
    tr[1] = TY[r];
    tr[2] = TZ[r];
  }
}

extern "C" void kernel_launch(void* const* d_in, const int* in_sizes, int n_in,
                              void* d_out, int out_size, void* d_ws, size_t ws_size,
                              hipStream_t stream) {
  const float* can = (const float*)d_in[0];          // canonical_vertices f32
  const float* def = (const float*)d_in[1];          // deformed_vertices  f32
  const int* nbr = (const int*)d_in[2];              // neighbors int32
  const unsigned char* msk = (const unsigned char*)d_in[3]; // bool mask (1B/elem)
  float* out = (float*)d_out;

  const dim3 grid(T_FRAMES * 5);   // 128 frames x 5 vertex chunks
  const dim3 block(256);           // 8 wave32s
  const size_t shmem = 2u * 15072u * sizeof(float);  // 120.6 KB (<320 KB/WGP)
  flame_se3_kernel<<<grid, block, shmem, stream>>>(can, def, nbr, msk, out);

  (void)in_sizes; (void)n_in; (void)out_size; (void)d_ws; (void)ws_size;
}